// PRGModule2_54262616818011
// MI455X (gfx1250) — compile-verified
//
#include <hip/hip_runtime.h>
#include <math.h>

typedef float v2f __attribute__((ext_vector_type(2)));
typedef float v8f __attribute__((ext_vector_type(8)));

#define BATCH 512
#define SEQ   79
#define DM    1024
#define ROWS  (BATCH * SEQ)   // 40448
#define NHEAD 8
#define C1    32

// gelu(x) = x * sigmoid(2*sqrt(2/pi)*(x + 0.044715 x^3))
// exponent constant: 2*log2(e)*sqrt(2/pi) = 2.3022082
__device__ __forceinline__ float fast_gelu(float x) {
    float u = x * (1.0f + 0.044715f * x * x);          // x + 0.044715 x^3
    float e = __builtin_amdgcn_exp2f(-2.3022082f * u); // v_exp_f32, branch-free
    return x * __builtin_amdgcn_rcpf(1.0f + e);        // v_rcp_f32
}

// ---------------------------------------------------------------------------
// Kernel 1: token projections. One wave32 per token row (coalesced float4
// reads of x, read exactly once). P[row][0..5]=piece_emb, [6]=color_emb, [7]=0
// ---------------------------------------------------------------------------
__global__ __launch_bounds__(256) void token_proj_kernel(
    const float* __restrict__ x,
    const float* __restrict__ piece_w, const float* __restrict__ piece_b,
    const float* __restrict__ color_w, const float* __restrict__ color_b,
    float* __restrict__ P)
{
    const int wave = threadIdx.x >> 5;
    const int lane = threadIdx.x & 31;
    const int row  = blockIdx.x * 8 + wave;   // grid is exact: ROWS/8 blocks

    const float4* xr = reinterpret_cast<const float4*>(x + (size_t)row * DM);

    float acc[7];
#pragma unroll
    for (int e = 0; e < 7; ++e) acc[e] = 0.0f;

#pragma unroll
    for (int it = 0; it < 8; ++it) {          // 8 iters * 32 lanes * 4 = 1024
        int idx = it * 32 + lane;
        float4 xv = xr[idx];
        int k = idx * 4;
        const float* pw = piece_w + (size_t)k * 6;
#pragma unroll
        for (int e = 0; e < 6; ++e)
            acc[e] += xv.x * pw[e] + xv.y * pw[6 + e] +
                      xv.z * pw[12 + e] + xv.w * pw[18 + e];
        const float* cw = color_w + k;
        acc[6] += xv.x * cw[0] + xv.y * cw[1] + xv.z * cw[2] + xv.w * cw[3];
    }
#pragma unroll
    for (int e = 0; e < 7; ++e)
        for (int off = 16; off > 0; off >>= 1)
            acc[e] += __shfl_xor(acc[e], off);

    if (lane == 0) {
        float* pr = P + (size_t)row * 8;
#pragma unroll
        for (int e = 0; e < 6; ++e) pr[e] = acc[e] + piece_b[e];
        pr[6] = acc[6] + color_b[0];
        pr[7] = 0.0f;
    }
}

// ---------------------------------------------------------------------------
// Kernel 2: pairwise MLP bias via WMMA. 4 blocks per batch element b; each
// block stages W1/W2/b1/b2 and per-token U,V,color in LDS. Waves own i-rows
// (outer) and sweep 5 j-tiles of 16 (inner, NOT unrolled to cap VGPRs),
// 8 chained V_WMMA_F32_16X16X4_F32 per tile over K=32.
// ---------------------------------------------------------------------------
__global__ __launch_bounds__(256) void pair_bias_kernel(
    const float* __restrict__ P,
    const float* __restrict__ w1, const float* __restrict__ b1,
    const float* __restrict__ w2, const float* __restrict__ b2,
    float* __restrict__ out)
{
    __shared__ float w1L[15 * C1];     // (15,32) row-major
    __shared__ float b1L[C1];
    __shared__ float w2L[C1 * NHEAD];  // (32,8) row-major
    __shared__ float b2L[NHEAD];
    __shared__ float PL[SEQ * 8];
    __shared__ float colorL[SEQ];
    __shared__ float UL[SEQ * C1];     // piece_i @ W1[2:8]
    __shared__ float VL[SEQ * C1];     // piece_j @ W1[8:14]

    const int tid  = threadIdx.x;
    const int b    = blockIdx.x >> 2;
    const int part = blockIdx.x & 3;

    for (int idx = tid; idx < 15 * C1; idx += 256) w1L[idx] = w1[idx];
    if (tid < C1) b1L[tid] = b1[tid];
    if (tid < C1 * NHEAD) w2L[tid] = w2[tid];
    if (tid < NHEAD) b2L[tid] = b2[tid];
    for (int idx = tid; idx < SEQ * 8; idx += 256)
        PL[idx] = P[(size_t)b * SEQ * 8 + idx];
    __syncthreads();

    for (int idx = tid; idx < SEQ; idx += 256) colorL[idx] = PL[idx * 8 + 6];
    for (int idx = tid; idx < SEQ * C1; idx += 256) {
        int j = idx >> 5, c = idx & 31;
        float u = 0.0f, v = 0.0f;
#pragma unroll
        for (int p = 0; p < 6; ++p) {
            float pv = PL[j * 8 + p];
            u += pv * w1L[(2 + p) * C1 + c];
            v += pv * w1L[(8 + p) * C1 + c];
        }
        UL[idx] = u;
        VL[idx] = v;
    }
    __syncthreads();

    const int wave = tid >> 5;
    const int lane = tid & 31;
    const int jl = lane & 15;     // A-matrix M-row (j in tile) / B-matrix column
    const int hi = lane >> 4;     // K-half selector for A/B fragments

    // Per-lane channel for slot k (k = 2t+s): c(k) = 4*(k>>1) + 2*hi + (k&1)
    // Hoist W1 rows 0,1,14 for this lane's 16 channels into VGPRs.
    float wA[16], wB[16], wC[16];
#pragma unroll
    for (int k = 0; k < 16; ++k) {
        int c = 4 * (k >> 1) + 2 * hi + (k & 1);
        wA[k] = w1L[c];                // W1 row 0  (dx)
        wB[k] = w1L[C1 + c];           // W1 row 1  (dy)
        wC[k] = w1L[14 * C1 + c];      // W1 row 14 (same_color)
    }

    // B fragments of W2 (32x8, columns padded to 16). 4x16 f32 layout:
    // lanes 0-15: VGPR0=K0,VGPR1=K1 ; lanes 16-31: VGPR0=K2,VGPR1=K3.
    v2f bf[8];
#pragma unroll
    for (int t = 0; t < 8; ++t) {
        int k0 = 4 * t + 2 * hi;
        v2f bv;
        bv.x = (jl < NHEAD) ? w2L[k0 * NHEAD + jl] : 0.0f;
        bv.y = (jl < NHEAD) ? w2L[(k0 + 1) * NHEAD + jl] : 0.0f;
        bf[t] = bv;
    }

    const int i0 = part * 20;
    const int ni = (SEQ - i0) < 20 ? (SEQ - i0) : 20;

    for (int i = i0 + wave; i < i0 + ni; i += 8) {
        // Hoist U[i,c] + b1[c] for this lane's channels.
        float su[16];
#pragma unroll
        for (int k = 0; k < 16; ++k) {
            int c = 4 * (k >> 1) + 2 * hi + (k & 1);
            su[k] = UL[i * C1 + c] + b1L[c];
        }
        float ci = colorL[i];
        int xi = (i < 64) ? (i >> 3) : 0, yi = (i < 64) ? (i & 7) : 0;

#pragma unroll 1   // keep VGPR allocation < 256; no benefit to unrolling
        for (int jt = 0; jt < 5; ++jt) {
            int j0 = jt * 16;
            int j  = j0 + jl;
            int jc = (j < SEQ) ? j : (SEQ - 1);  // clamp; masked on store

            float same = (ci == colorL[jc]) ? 1.0f : 0.0f;
            int xj = (jc < 64) ? (jc >> 3) : 0, yj = (jc < 64) ? (jc & 7) : 0;
            float dxf = (float)(xi - xj);
            float dyf = (float)(yi - yj);
            const float* Vr = VL + jc * C1;

            v8f acc = {};
#pragma unroll
            for (int t = 0; t < 8; ++t) {
                int c0 = 4 * t + 2 * hi;
                float pre0 = su[2 * t]     + Vr[c0]     + dxf * wA[2 * t]
                           + dyf * wB[2 * t]     + same * wC[2 * t];
                float pre1 = su[2 * t + 1] + Vr[c0 + 1] + dxf * wA[2 * t + 1]
                           + dyf * wB[2 * t + 1] + same * wC[2 * t + 1];
                v2f a; a.x = fast_gelu(pre0); a.y = fast_gelu(pre1);
                acc = __builtin_amdgcn_wmma_f32_16x16x4_f32(
                    false, a, false, bf[t], (short)0, acc, false, false);
            }

            // D layout: VGPR r -> (M = r + 8*hi, N = jl). M is j-in-tile, N head.
            int head = jl;
            if (head < NHEAD) {
                float bb = b2L[head];
                size_t base = (((size_t)b * NHEAD + head) * SEQ + i) * SEQ;
                int jb = j0 + 8 * hi;
#pragma unroll
                for (int r = 0; r < 8; ++r) {
                    int jj = jb + r;
                    if (jj < SEQ) out[base + jj] = acc[r] + bb;
                }
            }
        }
    }
}

// ---------------------------------------------------------------------------
extern "C" void kernel_launch(void* const* d_in, const int* in_sizes, int n_in,
                              void* d_out, int out_size, void* d_ws, size_t ws_size,
                              hipStream_t stream) {
    const float* x       = (const float*)d_in[0];
    const float* piece_w = (const float*)d_in[1];
    const float* piece_b = (const float*)d_in[2];
    const float* color_w = (const float*)d_in[3];
    const float* color_b = (const float*)d_in[4];
    const float* mlp1_w  = (const float*)d_in[5];
    const float* mlp1_b  = (const float*)d_in[6];
    const float* mlp2_w  = (const float*)d_in[7];
    const float* mlp2_b  = (const float*)d_in[8];

    float* P = (float*)d_ws;                  // (BATCH*SEQ, 8) = 1.29 MB

    token_proj_kernel<<<ROWS / 8, 256, 0, stream>>>(
        x, piece_w, piece_b, color_w, color_b, P);

    pair_bias_kernel<<<BATCH * 4, 256, 0, stream>>>(
        P, mlp1_w, mlp1_b, mlp2_w, mlp2_b, (float*)d_out);
}